// GCNFeatureSelector_47304769798459
// MI455X (gfx1250) — compile-verified
//
#include <hip/hip_runtime.h>

#define N_NODES 50000
#define N_EDGES 800000
#define IN_DIM  64
#define HID_DIM 128

typedef float v2f __attribute__((ext_vector_type(2)));
typedef float v4f __attribute__((ext_vector_type(4)));
typedef float v8f __attribute__((ext_vector_type(8)));

// ---------------------------------------------------------------------------
// Elementwise helpers
// ---------------------------------------------------------------------------
__global__ void gcn_zero_f32(float* __restrict__ p, int n) {
  int i = blockIdx.x * blockDim.x + threadIdx.x;
  if (i < n) p[i] = 0.0f;
}

__global__ void gcn_degree(const int* __restrict__ dst, float* __restrict__ deg, int E) {
  int e = blockIdx.x * blockDim.x + threadIdx.x;
  if (e < E) atomicAdd(&deg[dst[e]], 1.0f);
}

// deg -> dinv in place; +1.0 accounts for the self-loop, so deg > 0 always.
__global__ void gcn_dinv(float* __restrict__ deg, int n) {
  int i = blockIdx.x * blockDim.x + threadIdx.x;
  if (i < n) deg[i] = rsqrtf(deg[i] + 1.0f);
}

// ---------------------------------------------------------------------------
// fp32 GEMM on the matrix cores: C[M,N] = A[M,K] * B[K,N], row-major.
// K, N compile-time -> all addressing is shifts + immediate offsets.
//
// Block = (32, N/16): one 16-row strip of C per block, wave tn owns the
// 16x16 tile at column tn*16. A tile (16xK) and the whole B panel (KxN)
// are staged in LDS once per block with coalesced loads, then fragments
// come from ds_load:
//   A LDS stride K+4 words  -> half-waves (K pair offset 2) hit disjoint banks
//   B LDS stride N+8 words  -> half-wave bank offset = 2*(N+8) mod 64 = 16
// WMMA fragment layout (V_WMMA_F32_16X16X4_F32):
//   A 16x4: lanes 0-15 M=lane K={k0,k0+1}; lanes 16-31 M=lane-16 K={k0+2,k0+3}
//   B 4x16: lane%16 = column, half-wave picks the K pair
//   C/D:    VGPR r -> row r + 8*(lane>=16), col lane%16
// ---------------------------------------------------------------------------
template <int K, int N>
__global__ __launch_bounds__(32 * (N / 16)) void gcn_gemm_wmma(
    const float* __restrict__ A, const float* __restrict__ B,
    float* __restrict__ C, int M) {
  constexpr int AS = K + 4;   // padded A row stride (words)
  constexpr int BS = N + 8;   // padded B row stride (words)
  __shared__ float Alds[16 * AS];
  __shared__ float Blds[K * BS];

  const int lane = threadIdx.x;              // 0..31
  const int tn   = threadIdx.y;              // 0..N/16-1
  const int tm   = blockIdx.x;
  const int tid  = tn * 32 + lane;
  constexpr int NT = 32 * (N / 16);

  // Stage A tile (16 x K) -- coalesced along K.
  const float* __restrict__ ag = A + (size_t)tm * 16 * K;
#pragma unroll
  for (int idx = tid; idx < 16 * K; idx += NT) {
    int r = idx / K, c = idx % K;            // K is a power of two
    Alds[r * AS + c] = ag[idx];
  }
  // Stage full B panel (K x N) -- coalesced.
#pragma unroll
  for (int idx = tid; idx < K * N; idx += NT) {
    int r = idx / N, c = idx % N;
    Blds[r * BS + c] = B[idx];
  }
  __syncthreads();

  const int half = lane >> 4;
  const int l16  = lane & 15;
  const float* __restrict__ ap = Alds + l16 * AS;        // even word offset
  const float* __restrict__ bp = Blds + tn * 16 + l16;

  v8f acc = {};
#pragma unroll
  for (int k0 = 0; k0 < K; k0 += 4) {
    const int kb = k0 + half * 2;
    v2f a = *(const v2f*)(ap + kb);          // ds_load_b64, 8B aligned
    v2f b;
    b.x = bp[kb * BS];
    b.y = bp[(kb + 1) * BS];
    acc = __builtin_amdgcn_wmma_f32_16x16x4_f32(
        false, a, false, b, (short)0, acc, false, false);
  }

  float* __restrict__ crow = C + ((size_t)tm * 16 + half * 8) * N + tn * 16 + l16;
#pragma unroll
  for (int r = 0; r < 8; ++r) crow[(size_t)r * N] = acc[r];
}

// ---------------------------------------------------------------------------
// Edge scatter: out[dst] += h[src] * dinv[src]*dinv[dst].
// 32 lanes per edge; lane owns D/32 consecutive columns -> one b128/b64
// gather per lane, then per-element global_atomic_add_f32. h rows and the
// accumulation target both fit in the 192MB L2.
// ---------------------------------------------------------------------------
template <int D>
__global__ void gcn_scatter(const int* __restrict__ src, const int* __restrict__ dst,
                            const float* __restrict__ h, const float* __restrict__ dinv,
                            float* __restrict__ out, int E) {
  constexpr int VW = D / 32;                 // 4 (D=128) or 2 (D=64)
  typedef float vload __attribute__((ext_vector_type(VW)));
  int t = blockIdx.x * blockDim.x + threadIdx.x;
  int e = t >> 5;
  int lane = t & 31;
  if (e >= E) return;
  int s = src[e];
  int d = dst[e];
  float w = dinv[s] * dinv[d];
  const float* __restrict__ hs = h + (size_t)s * D + lane * VW;
  float* __restrict__ od = out + (size_t)d * D + lane * VW;
  vload hv = *(const vload*)hs;              // 16B/8B aligned vector load
#pragma unroll
  for (int j = 0; j < VW; ++j) atomicAdd(&od[j], hv[j] * w);
}

// Layer-1 finish: agg <- relu(agg + h*dinv^2 + b)   (self-loop folded in)
__global__ void gcn_finish1(float* __restrict__ agg, const float* __restrict__ h,
                            const float* __restrict__ dinv, const float* __restrict__ b,
                            int n) {
  int i = blockIdx.x * blockDim.x + threadIdx.x;
  if (i >= n) return;
  int node = i >> 7;           // / HID_DIM
  int j    = i & (HID_DIM - 1);
  float di = dinv[node];
  float v  = agg[i] + h[i] * di * di + b[j];
  agg[i] = v > 0.0f ? v : 0.0f;
}

// Layer-2 init: out <- h*dinv^2 + b  (writes every output element; scatter
// then adds neighbor terms atomically)
__global__ void gcn_init_out(float* __restrict__ out, const float* __restrict__ h,
                             const float* __restrict__ dinv, const float* __restrict__ b,
                             int n) {
  int i = blockIdx.x * blockDim.x + threadIdx.x;
  if (i >= n) return;
  int node = i >> 6;           // / IN_DIM
  int j    = i & (IN_DIM - 1);
  float di = dinv[node];
  out[i] = h[i] * di * di + b[j];
}

// ---------------------------------------------------------------------------
// Launch
// ---------------------------------------------------------------------------
extern "C" void kernel_launch(void* const* d_in, const int* in_sizes, int n_in,
                              void* d_out, int out_size, void* d_ws, size_t ws_size,
                              hipStream_t stream) {
  const float* x  = (const float*)d_in[0];          // [N, 64]
  const int*   ei = (const int*)d_in[1];            // [2, E]
  const float* W1 = (const float*)d_in[2];          // [64, 128]
  const float* b1 = (const float*)d_in[3];          // [128]
  const float* W2 = (const float*)d_in[4];          // [128, 64]
  const float* b2 = (const float*)d_in[5];          // [64]
  float* out = (float*)d_out;                       // [N, 64]

  const int* src = ei;
  const int* dst = ei + N_EDGES;

  // Workspace: dinv (0.2MB) | h1 (25.6MB, reused as h2) | agg1 (25.6MB)
  char* ws = (char*)d_ws;
  float* dinv = (float*)ws;
  size_t off  = (((size_t)N_NODES * 4) + 255) & ~(size_t)255;
  float* h1   = (float*)(ws + off);
  float* agg1 = h1 + (size_t)N_NODES * HID_DIM;
  float* h2   = h1;  // h1 dead after finish1; reuse for the layer-2 GEMM

  const int TB = 256;

  // 1) degrees -> dinv
  gcn_zero_f32<<<(N_NODES + TB - 1) / TB, TB, 0, stream>>>(dinv, N_NODES);
  gcn_degree<<<(N_EDGES + TB - 1) / TB, TB, 0, stream>>>(dst, dinv, N_EDGES);
  gcn_dinv<<<(N_NODES + TB - 1) / TB, TB, 0, stream>>>(dinv, N_NODES);

  // 2) h1 = x @ W1  (50000x64 * 64x128)
  gcn_gemm_wmma<IN_DIM, HID_DIM>
      <<<dim3(N_NODES / 16), dim3(32, HID_DIM / 16), 0, stream>>>(x, W1, h1, N_NODES);

  // 3) agg1 = scatter-add of normalized h1 over edges, then bias+relu
  {
    int n = N_NODES * HID_DIM;
    gcn_zero_f32<<<(n + TB - 1) / TB, TB, 0, stream>>>(agg1, n);
    long threads = (long)N_EDGES * 32;
    gcn_scatter<HID_DIM><<<(threads + TB - 1) / TB, TB, 0, stream>>>(
        src, dst, h1, dinv, agg1, N_EDGES);
    gcn_finish1<<<(n + TB - 1) / TB, TB, 0, stream>>>(agg1, h1, dinv, b1, n);
  }

  // 4) h2 = hidden @ W2  (50000x128 * 128x64)
  gcn_gemm_wmma<HID_DIM, IN_DIM>
      <<<dim3(N_NODES / 16), dim3(32, IN_DIM / 16), 0, stream>>>(agg1, W2, h2, N_NODES);

  // 5) out = h2*dinv^2 + b2, then scatter-add edge messages
  {
    int n = N_NODES * IN_DIM;
    gcn_init_out<<<(n + TB - 1) / TB, TB, 0, stream>>>(out, h2, dinv, b2, n);
    long threads = (long)N_EDGES * 32;
    gcn_scatter<IN_DIM><<<(threads + TB - 1) / TB, TB, 0, stream>>>(
        src, dst, h2, dinv, out, N_EDGES);
  }
}